// GAT_38560216384098
// MI455X (gfx1250) — compile-verified
//
#include <hip/hip_runtime.h>
#include <hip/hip_bf16.h>

#define IN_DIM 256
#define HEADS 4
#define OUT_C 64
#define HC 256            // HEADS*OUT_C
#define NEG_SLOPE 0.2f

typedef float v2f __attribute__((ext_vector_type(2)));
typedef float v8f __attribute__((ext_vector_type(8)));

// ---------------------------------------------------------------------------
// Kernel 1: fused projection GEMMs via fp32 WMMA (16x16x4).
//   xl = x @ W_l + b_l ; xr = x @ W_r + b_r
// grid.x = ceil(N/16) M-tiles; block = 256 threads = 8 waves.
// Wave w computes column tiles {w, w+8, w+16, w+24} of the fused 512-wide
// [W_l | W_r] output. A tile (16x256 of x) is staged in LDS once per block.
// ---------------------------------------------------------------------------
__global__ __launch_bounds__(256)
void gat_proj_wmma(const float* __restrict__ x,
                   const float* __restrict__ Wl, const float* __restrict__ bl,
                   const float* __restrict__ Wr, const float* __restrict__ br,
                   float* __restrict__ xl, float* __restrict__ xr, int nnodes)
{
    __shared__ float As[16 * 260];        // padded stride: avoid bank conflicts
    const int tid  = threadIdx.x;
    const int wave = tid >> 5;
    const int lane = tid & 31;
    const int m0   = blockIdx.x * 16;

    for (int i = tid; i < 16 * 256; i += 256) {   // coalesced A stage
        int r = i >> 8, c = i & 255;
        int gr = m0 + r;
        As[r * 260 + c] = (gr < nnodes) ? x[gr * IN_DIM + c] : 0.0f;
    }
    __syncthreads();

    v8f acc[4] = {};                       // 4 column tiles per wave
    const int lrow = lane & 15;
    const int hi   = lane >> 4;            // half-wave select (0/1)

    #pragma unroll 4
    for (int k0 = 0; k0 < IN_DIM; k0 += 4) {
        const int kk = k0 + (hi << 1);     // ISA A/B layout: K = k0(+2) per half
        v2f a;
        a.x = As[lrow * 260 + kk];
        a.y = As[lrow * 260 + kk + 1];
        #pragma unroll
        for (int j = 0; j < 4; ++j) {
            const int n0 = (wave + (j << 3)) << 4;       // 0..496
            const float* W = (n0 < HC) ? Wl : Wr;
            const int c = (n0 & (HC - 1)) + lrow;
            v2f b;
            b.x = W[kk * HC + c];
            b.y = W[(kk + 1) * HC + c];
            acc[j] = __builtin_amdgcn_wmma_f32_16x16x4_f32(
                false, a, false, b, (short)0, acc[j], false, false);
        }
    }

    #pragma unroll
    for (int j = 0; j < 4; ++j) {
        const int n0 = (wave + (j << 3)) << 4;
        float* o = (n0 < HC) ? xl : xr;
        const float* bias = (n0 < HC) ? bl : br;
        const int c = (n0 & (HC - 1)) + lrow;
        const float bv = bias[c];
        #pragma unroll
        for (int v = 0; v < 8; ++v) {      // C/D layout: M = v + 8*hi, N = lane&15
            const int r = m0 + v + (hi << 3);
            if (r < nnodes) o[r * HC + c] = acc[j][v] + bv;
        }
    }
}

// ---------------------------------------------------------------------------
// Float atomic max via ordered-int trick (works with -inf init).
// ---------------------------------------------------------------------------
__device__ inline void atomicMaxF(float* addr, float val) {
    if (val >= 0.0f) atomicMax((int*)addr, __float_as_int(val));
    else             atomicMin((unsigned int*)addr, __float_as_uint(val));
}

// ---------------------------------------------------------------------------
// Kernel 0: out = bias (broadcast), mx = -inf, dn = 0
// ---------------------------------------------------------------------------
__global__ void gat_init(float* __restrict__ out, const float* __restrict__ bias,
                         float* __restrict__ mx, float* __restrict__ dn,
                         int nout, int nnh)
{
    int t = blockIdx.x * blockDim.x + threadIdx.x;
    if (t < nout) out[t] = bias[t & (HC - 1)];
    if (t < nnh) { mx[t] = -__builtin_huge_valf(); dn[t] = 0.0f; }
}

// ---------------------------------------------------------------------------
// Kernel 2: per-edge attention logits + segment max.
// One wave32 per edge; lane owns 8 channels (one head per 8-lane group).
// ---------------------------------------------------------------------------
__global__ __launch_bounds__(256)
void gat_logits(const int* __restrict__ ei, const float* __restrict__ xl,
                const float* __restrict__ xr, const float* __restrict__ att,
                float* __restrict__ lg, float* __restrict__ mx,
                int E, int Etot)
{
    int e = (blockIdx.x * blockDim.x + threadIdx.x) >> 5;
    int lane = threadIdx.x & 31;
    if (e >= Etot) return;
    int src, dst;
    if (e < E) { src = ei[e]; dst = ei[E + e]; }
    else       { src = dst = e - E; }                 // self loop

    const float4* pl = (const float4*)(xl + (size_t)src * HC + lane * 8);
    const float4* pr = (const float4*)(xr + (size_t)dst * HC + lane * 8);
    const float4* pa = (const float4*)(att + lane * 8);
    float s = 0.0f;
    #pragma unroll
    for (int q = 0; q < 2; ++q) {
        float4 l = pl[q], r = pr[q], a = pa[q];
        float v;
        v = l.x + r.x; s += a.x * (v > 0.0f ? v : NEG_SLOPE * v);
        v = l.y + r.y; s += a.y * (v > 0.0f ? v : NEG_SLOPE * v);
        v = l.z + r.z; s += a.z * (v > 0.0f ? v : NEG_SLOPE * v);
        v = l.w + r.w; s += a.w * (v > 0.0f ? v : NEG_SLOPE * v);
    }
    s += __shfl_xor(s, 1, 32);                        // reduce 8 lanes -> head
    s += __shfl_xor(s, 2, 32);
    s += __shfl_xor(s, 4, 32);
    if ((lane & 7) == 0) {
        int h = lane >> 3;
        lg[e * HEADS + h] = s;
        atomicMaxF(&mx[dst * HEADS + h], s);
    }
}

// ---------------------------------------------------------------------------
// Kernel 3: ex = exp(logit - max[dst]); denom[dst] += ex. One thread per (e,h).
// ---------------------------------------------------------------------------
__global__ void gat_expsum(const int* __restrict__ ei, const float* __restrict__ mx,
                           float* __restrict__ lg, float* __restrict__ dn,
                           int E, int Etot)
{
    int idx = blockIdx.x * blockDim.x + threadIdx.x;
    if (idx >= Etot * HEADS) return;
    int e = idx >> 2, h = idx & 3;
    int dst = (e < E) ? ei[E + e] : (e - E);
    float v = __expf(lg[idx] - mx[dst * HEADS + h]);
    lg[idx] = v;
    atomicAdd(&dn[dst * HEADS + h], v);
}

// ---------------------------------------------------------------------------
// Kernel 4: out[dst] += (ex/denom[dst]) * xl[src]. One wave32 per edge.
// ---------------------------------------------------------------------------
__global__ __launch_bounds__(256)
void gat_aggregate(const int* __restrict__ ei, const float* __restrict__ xl,
                   const float* __restrict__ lg, const float* __restrict__ dn,
                   float* __restrict__ out, int E, int Etot)
{
    int e = (blockIdx.x * blockDim.x + threadIdx.x) >> 5;
    int lane = threadIdx.x & 31;
    if (e >= Etot) return;
    int src, dst;
    if (e < E) { src = ei[e]; dst = ei[E + e]; }
    else       { src = dst = e - E; }
    int h = lane >> 3;
    float alpha = lg[e * HEADS + h] / dn[dst * HEADS + h];
    const float* pl = xl + (size_t)src * HC + lane * 8;
    float* po = out + (size_t)dst * HC + lane * 8;
    #pragma unroll
    for (int i = 0; i < 8; ++i)
        atomicAdd(&po[i], alpha * pl[i]);
}

// ---------------------------------------------------------------------------
extern "C" void kernel_launch(void* const* d_in, const int* in_sizes, int n_in,
                              void* d_out, int out_size, void* d_ws, size_t ws_size,
                              hipStream_t stream)
{
    const float* x    = (const float*)d_in[0];
    const int*   ei   = (const int*)  d_in[1];
    const float* Wl   = (const float*)d_in[2];
    const float* bl   = (const float*)d_in[3];
    const float* Wr   = (const float*)d_in[4];
    const float* br   = (const float*)d_in[5];
    const float* att  = (const float*)d_in[6];
    const float* bias = (const float*)d_in[7];

    const int N    = in_sizes[0] / IN_DIM;
    const int E    = in_sizes[1] / 2;
    const int Etot = E + N;                       // + self loops

    float* ws = (float*)d_ws;
    float* xl = ws;                               // N*HC
    float* xr = xl + (size_t)N * HC;              // N*HC
    float* mx = xr + (size_t)N * HC;              // N*HEADS
    float* dn = mx + (size_t)N * HEADS;           // N*HEADS
    float* lg = dn + (size_t)N * HEADS;           // Etot*HEADS
    float* out = (float*)d_out;

    // 1) fused projections (WMMA f32)
    gat_proj_wmma<<<(N + 15) / 16, 256, 0, stream>>>(x, Wl, bl, Wr, br, xl, xr, N);

    // 2) init out=bias, mx=-inf, dn=0
    int initN = N * HC;
    gat_init<<<(initN + 255) / 256, 256, 0, stream>>>(out, bias, mx, dn, initN, N * HEADS);

    // 3) logits + segment max
    long long lt = (long long)Etot * 32;
    gat_logits<<<(int)((lt + 255) / 256), 256, 0, stream>>>(ei, xl, xr, att, lg, mx, E, Etot);

    // 4) exp + segment sum
    int en = Etot * HEADS;
    gat_expsum<<<(en + 255) / 256, 256, 0, stream>>>(ei, mx, lg, dn, E, Etot);

    // 5) weighted scatter-aggregation
    gat_aggregate<<<(int)((lt + 255) / 256), 256, 0, stream>>>(ei, xl, lg, dn, out, E, Etot);
}